// VoxelBackBone8x_53360673686080
// MI455X (gfx1250) — compile-verified
//
#include <hip/hip_runtime.h>
#include <hip/hip_bf16.h>
#include <math.h>

typedef __attribute__((ext_vector_type(2))) float v2f;
typedef __attribute__((ext_vector_type(8))) float v8f;

// ---- grid dims per level ----
static constexpr int D1=41, H1=160, W1=128;   static constexpr int V1=D1*H1*W1;   // 839680
static constexpr int D2=21, H2=80,  W2=64;    static constexpr int V2=D2*H2*W2;   // 107520
static constexpr int D3=11, H3=40,  W3=32;    static constexpr int V3=D3*H3*W3;   // 14080
static constexpr int D4=5,  H4=20,  W4=16;    static constexpr int V4=D4*H4*W4;   // 1600
static constexpr int D5=2,  H5=20,  W5=16;    static constexpr int V5=D5*H5*W5;   // 640

// ------------------------------------------------------------------ utilities
__global__ void fill_i32(int* __restrict__ p, int n, int v) {
    int i = blockIdx.x * 256 + threadIdx.x;
    if (i < n) p[i] = v;
}
__global__ void fill_f32(float* __restrict__ p, int n) {
    int i = blockIdx.x * 256 + threadIdx.x;
    if (i < n) p[i] = 0.0f;
}
__global__ void init_counters(int* __restrict__ cnt, int n0) {
    if (threadIdx.x == 0) {
        cnt[0] = n0;
        for (int i = 1; i < 8; ++i) cnt[i] = 0;
    }
}

// Scatter input voxels: build level-1 index grid, coord list, and 8-ch
// feature rows with channels 4..6 zeroed (feats.at[:,4:7].set(0)).
__global__ void scatter_input(const float* __restrict__ vf, const int* __restrict__ vc,
                              int N, int* __restrict__ idx1, int* __restrict__ coords1,
                              float* __restrict__ f0) {
    int i = blockIdx.x * 256 + threadIdx.x;
    if (i >= N) return;
    int zi = vc[i * 4 + 1], yi = vc[i * 4 + 2], xi = vc[i * 4 + 3];
    int lin = (zi * H1 + yi) * W1 + xi;
    idx1[lin]  = i;
    coords1[i] = lin;
    #pragma unroll
    for (int c = 0; c < 8; ++c) {
        float v = vf[i * 8 + c];
        f0[i * 8 + c] = (c >= 4 && c < 7) ? 0.0f : v;
    }
}

// Enumerate active output voxels of a strided sparse conv (down_mask):
// output voxel is active iff any input voxel in its kernel footprint is active.
__global__ void enumerate_down(const int* __restrict__ idxIn, int inD, int inH, int inW,
                               int outD, int outH, int outW,
                               int kd, int kh, int kw, int sz, int sy, int sx,
                               int pz, int py, int px,
                               int* __restrict__ idxOut, int* __restrict__ coordsOut,
                               int* __restrict__ cntOut) {
    int o = blockIdx.x * 256 + threadIdx.x;
    int tot = outD * outH * outW;
    if (o >= tot) return;
    int x = o % outW, y = (o / outW) % outH, z = o / (outW * outH);
    bool act = false;
    for (int a = 0; a < kd; ++a) {
        int zi = z * sz + a - pz;
        if (zi < 0 || zi >= inD) continue;
        for (int b = 0; b < kh; ++b) {
            int yi = y * sy + b - py;
            if (yi < 0 || yi >= inH) continue;
            for (int c = 0; c < kw; ++c) {
                int xi = x * sx + c - px;
                if (xi < 0 || xi >= inW) continue;
                if (idxIn[(zi * inH + yi) * inW + xi] >= 0) act = true;
            }
        }
    }
    if (act) {
        int r = atomicAdd(cntOut, 1);
        idxOut[o]    = r;
        coordsOut[r] = o;
    }
}

// Build per-row neighbor table (gather indices into the input row space).
// Works for submanifold (stride 1, in==out grid) and strided down convs.
__global__ void build_nbr(const int* __restrict__ coords, const int* __restrict__ cntPtr,
                          const int* __restrict__ idxIn, int inD, int inH, int inW,
                          int outH, int outW,
                          int kd, int kh, int kw, int sz, int sy, int sx,
                          int pz, int py, int px, int* __restrict__ nbr) {
    int r = blockIdx.x * 256 + threadIdx.x;
    if (r >= cntPtr[0]) return;
    int lin = coords[r];
    int x = lin % outW, y = (lin / outW) % outH, z = lin / (outW * outH);
    int t = 0;
    for (int a = 0; a < kd; ++a)
        for (int b = 0; b < kh; ++b)
            for (int c = 0; c < kw; ++c, ++t) {
                int zi = z * sz + a - pz, yi = y * sy + b - py, xi = x * sx + c - px;
                int v = -1;
                if (zi >= 0 && zi < inD && yi >= 0 && yi < inH && xi >= 0 && xi < inW)
                    v = idxIn[(zi * inH + yi) * inW + xi];
                nbr[r * 27 + t] = v;
            }
}

// ------------------------------------------------------------------ WMMA conv
// 8 waves (256 threads) per block on one WGP. Each wave32 owns a 16(row)x16(outc)
// f32 tile computed with V_WMMA_F32_16X16X4_F32. All 8 waves share one out-channel
// tile; the per-tap weight slice W[t][:, ctile:ctile+16] (<= 8 KB) is staged into
// LDS once per tap with coalesced loads, so B fragments come from LDS broadcasts
// instead of 8x-redundant strided global loads. A[m][k] is the gathered neighbor
// feature (irregular; served from global/L2). Epilogue: BN(+residual)+ReLU, fp32.
#define CONV_WAVES 8
__global__ __launch_bounds__(32 * CONV_WAVES)
void conv_wmma(const float* __restrict__ fin, const int* __restrict__ nbr,
               const float* __restrict__ Wt, const float* __restrict__ gamma,
               const float* __restrict__ beta, const float* __restrict__ res,
               float* __restrict__ fout, const int* __restrict__ cntPtr,
               int cin, int cout, int ntaps) {
    __shared__ float sw[128 * 16];                // max cin(128) x 16 out-ch, 8 KB

    const int count = cntPtr[0];
    if ((int)blockIdx.x * (16 * CONV_WAVES) >= count) return;   // uniform whole-block exit

    const int tid   = threadIdx.x;
    const int wave  = tid >> 5;
    const int lane  = tid & 31;
    const int lm    = lane & 15;
    const int half  = lane >> 4;                  // 0: K {0,1} | 1: K {2,3} per step
    const int rowBase = blockIdx.x * (16 * CONV_WAVES) + wave * 16;
    const int ctile   = blockIdx.y * 16;
    const int m     = rowBase + lm;               // A row (M dim)
    const bool mval = (m < count);
    const int n     = ctile + lm;                 // B/C column (N dim)
    const int nw    = cin * 16;                   // weight-tile elements per tap

    v8f acc = {};
    for (int t = 0; t < ntaps; ++t) {
        // ---- stage W[t][:, ctile:ctile+16] into LDS (coalesced, cooperative) ----
        __syncthreads();                          // protect previous tap's reads
        for (int i = tid; i < nw; i += 32 * CONV_WAVES) {
            int ci = i >> 4, cc = i & 15;
            sw[i] = Wt[((size_t)t * cin + ci) * cout + (ctile + cc)];
        }
        __syncthreads();

        // ---- gather A rows + WMMA over K in steps of 4 ----
        int rA = mval ? nbr[m * 27 + t] : -1;
        const float* arow = (rA >= 0) ? (fin + (size_t)rA * cin) : nullptr;
        for (int k0 = 0; k0 < cin; k0 += 4) {
            const int kk = k0 + 2 * half;
            v2f a;
            if (arow) a = *(const v2f*)(arow + kk);     // 8B-aligned (cin mult of 8)
            else      { a.x = 0.0f; a.y = 0.0f; }
            v2f b;
            b.x = sw[(kk + 0) * 16 + lm];
            b.y = sw[(kk + 1) * 16 + lm];
            // D = A(16x4) * B(4x16) + C : fp32 throughout (matches reference numerics)
            acc = __builtin_amdgcn_wmma_f32_16x16x4_f32(false, a, false, b,
                                                        (short)0, acc, false, false);
        }
    }

    // Epilogue: eval-mode BN (mean 0, var 1): x * g * rsqrt(1+eps) + b, +res, ReLU
    const float rs = rsqrtf(1.0f + 1e-3f);
    const float sc = gamma[n] * rs;
    const float sb = beta[n];
    #pragma unroll
    for (int v = 0; v < 8; ++v) {
        int row = rowBase + v + 8 * half;         // C/D layout: VGPR v -> M = v+8*half
        if (row < count) {
            float val = acc[v] * sc + sb;
            if (res) val += res[(size_t)row * cout + n];
            fout[(size_t)row * cout + n] = fmaxf(val, 0.0f);
        }
    }
}

// Scatter final active rows to the dense (1,2,20,16,128) NDHWC output.
__global__ void scatter_output(const float* __restrict__ f5, const int* __restrict__ coords5,
                               const int* __restrict__ cntPtr, float* __restrict__ out) {
    int r = blockIdx.x;
    if (r >= cntPtr[0]) return;
    int lin = coords5[r];                         // (z*H5 + y)*W5 + x
    int c = threadIdx.x;                          // 128 channels
    out[(size_t)lin * 128 + c] = f5[r * 128 + c];
}

// ------------------------------------------------------------------ driver
extern "C" void kernel_launch(void* const* d_in, const int* in_sizes, int n_in,
                              void* d_out, int out_size, void* d_ws, size_t ws_size,
                              hipStream_t stream) {
    (void)n_in; (void)ws_size;
    const float* vfeat  = (const float*)d_in[0];
    const int*   vcoord = (const int*)d_in[1];
    const float *Wt[15], *Ga[15], *Be[15];
    for (int i = 0; i < 15; ++i) {
        Wt[i] = (const float*)d_in[2 + i];
        Ga[i] = (const float*)d_in[17 + i];
        Be[i] = (const float*)d_in[32 + i];
    }
    const int N = in_sizes[0] / 8;

    // ---- carve workspace ----
    char* ws = (char*)d_ws;
    size_t off = 0;
    auto alloc = [&](size_t bytes) -> void* {
        void* p = ws + off;
        off = (off + bytes + 255) & ~(size_t)255;
        return p;
    };
    int*   cnt     = (int*)alloc(8 * sizeof(int));           // [0]=N1 [1..4]=N2..N5
    int*   idx1    = (int*)alloc((size_t)V1 * 4);
    int*   idx2    = (int*)alloc((size_t)V2 * 4);
    int*   idx3    = (int*)alloc((size_t)V3 * 4);
    int*   idx4    = (int*)alloc((size_t)V4 * 4);
    int*   idx5    = (int*)alloc((size_t)V5 * 4);
    int*   coords1 = (int*)alloc((size_t)N  * 4);
    int*   coords2 = (int*)alloc((size_t)V2 * 4);
    int*   coords3 = (int*)alloc((size_t)V3 * 4);
    int*   coords4 = (int*)alloc((size_t)V4 * 4);
    int*   coords5 = (int*)alloc((size_t)V5 * 4);
    size_t nbrRows = (size_t)(N > V2 ? N : V2);
    int*   nbr     = (int*)alloc(nbrRows * 27 * 4);          // shared, rebuilt per stage
    float* f0      = (float*)alloc((size_t)N  * 8   * 4);
    float* f1a     = (float*)alloc((size_t)N  * 16  * 4);
    float* f1b     = (float*)alloc((size_t)N  * 16  * 4);
    float* f1c     = (float*)alloc((size_t)N  * 16  * 4);
    float* f2a     = (float*)alloc((size_t)V2 * 32  * 4);
    float* f2b     = (float*)alloc((size_t)V2 * 32  * 4);
    float* f3a     = (float*)alloc((size_t)V3 * 64  * 4);
    float* f3b     = (float*)alloc((size_t)V3 * 64  * 4);
    float* f4a     = (float*)alloc((size_t)V4 * 128 * 4);
    float* f4b     = (float*)alloc((size_t)V4 * 128 * 4);
    float* f5      = (float*)alloc((size_t)V5 * 128 * 4);

    const int TB = 256;
    auto blks = [&](int n) { return (n + TB - 1) / TB; };

    // ---- init ----
    fill_i32<<<blks(V1), TB, 0, stream>>>(idx1, V1, -1);
    fill_i32<<<blks(V2), TB, 0, stream>>>(idx2, V2, -1);
    fill_i32<<<blks(V3), TB, 0, stream>>>(idx3, V3, -1);
    fill_i32<<<blks(V4), TB, 0, stream>>>(idx4, V4, -1);
    fill_i32<<<blks(V5), TB, 0, stream>>>(idx5, V5, -1);
    fill_f32<<<blks(out_size), TB, 0, stream>>>((float*)d_out, out_size);
    init_counters<<<1, 32, 0, stream>>>(cnt, N);
    scatter_input<<<blks(N), TB, 0, stream>>>(vfeat, vcoord, N, idx1, coords1, f0);

    auto conv = [&](const float* fin, int li, const float* res, float* fout,
                    const int* cptr, int maxR, int cin, int cout, int ntaps) {
        dim3 g((maxR + 16 * CONV_WAVES - 1) / (16 * CONV_WAVES), cout / 16);
        conv_wmma<<<g, 32 * CONV_WAVES, 0, stream>>>(fin, nbr, Wt[li], Ga[li], Be[li],
                                                     res, fout, cptr, cin, cout, ntaps);
    };

    // ---- stage 1: conv_input + two residual blocks (submanifold, full res) ----
    build_nbr<<<blks(N), TB, 0, stream>>>(coords1, cnt + 0, idx1, D1, H1, W1, H1, W1,
                                          3, 3, 3, 1, 1, 1, 1, 1, 1, nbr);
    conv(f0,  0,  nullptr, f1a, cnt + 0, N, 8,  16, 27);     // conv_input
    conv(f1a, 1,  nullptr, f1b, cnt + 0, N, 16, 16, 27);     // block1 conv1
    conv(f1b, 2,  f1a,     f1c, cnt + 0, N, 16, 16, 27);     // block1 conv2 + res
    conv(f1c, 3,  nullptr, f1a, cnt + 0, N, 16, 16, 27);     // block2 conv1
    conv(f1a, 4,  f1c,     f1b, cnt + 0, N, 16, 16, 27);     // block2 conv2 + res -> x1=f1b

    // ---- stage 2: stride-2 down + 2 subm (32 ch) ----
    enumerate_down<<<blks(V2), TB, 0, stream>>>(idx1, D1, H1, W1, D2, H2, W2,
                                                3, 3, 3, 2, 2, 2, 1, 1, 1,
                                                idx2, coords2, cnt + 1);
    build_nbr<<<blks(V2), TB, 0, stream>>>(coords2, cnt + 1, idx1, D1, H1, W1, H2, W2,
                                           3, 3, 3, 2, 2, 2, 1, 1, 1, nbr);
    conv(f1b, 5,  nullptr, f2a, cnt + 1, V2, 16, 32, 27);
    build_nbr<<<blks(V2), TB, 0, stream>>>(coords2, cnt + 1, idx2, D2, H2, W2, H2, W2,
                                           3, 3, 3, 1, 1, 1, 1, 1, 1, nbr);
    conv(f2a, 6,  nullptr, f2b, cnt + 1, V2, 32, 32, 27);
    conv(f2b, 7,  nullptr, f2a, cnt + 1, V2, 32, 32, 27);    // x2 = f2a

    // ---- stage 3: stride-2 down + 2 subm (64 ch) ----
    enumerate_down<<<blks(V3), TB, 0, stream>>>(idx2, D2, H2, W2, D3, H3, W3,
                                                3, 3, 3, 2, 2, 2, 1, 1, 1,
                                                idx3, coords3, cnt + 2);
    build_nbr<<<blks(V3), TB, 0, stream>>>(coords3, cnt + 2, idx2, D2, H2, W2, H3, W3,
                                           3, 3, 3, 2, 2, 2, 1, 1, 1, nbr);
    conv(f2a, 8,  nullptr, f3a, cnt + 2, V3, 32, 64, 27);
    build_nbr<<<blks(V3), TB, 0, stream>>>(coords3, cnt + 2, idx3, D3, H3, W3, H3, W3,
                                           3, 3, 3, 1, 1, 1, 1, 1, 1, nbr);
    conv(f3a, 9,  nullptr, f3b, cnt + 2, V3, 64, 64, 27);
    conv(f3b, 10, nullptr, f3a, cnt + 2, V3, 64, 64, 27);    // x3 = f3a

    // ---- stage 4: stride-2 down, pad (0,1,1) + 2 subm (128 ch) ----
    enumerate_down<<<blks(V4), TB, 0, stream>>>(idx3, D3, H3, W3, D4, H4, W4,
                                                3, 3, 3, 2, 2, 2, 0, 1, 1,
                                                idx4, coords4, cnt + 3);
    build_nbr<<<blks(V4), TB, 0, stream>>>(coords4, cnt + 3, idx3, D3, H3, W3, H4, W4,
                                           3, 3, 3, 2, 2, 2, 0, 1, 1, nbr);
    conv(f3a, 11, nullptr, f4a, cnt + 3, V4, 64, 128, 27);
    build_nbr<<<blks(V4), TB, 0, stream>>>(coords4, cnt + 3, idx4, D4, H4, W4, H4, W4,
                                           3, 3, 3, 1, 1, 1, 1, 1, 1, nbr);
    conv(f4a, 12, nullptr, f4b, cnt + 3, V4, 128, 128, 27);
    conv(f4b, 13, nullptr, f4a, cnt + 3, V4, 128, 128, 27);  // x4 = f4a

    // ---- conv_out: kernel (3,1,1), stride (2,1,1), pad 0 ----
    enumerate_down<<<blks(V5), TB, 0, stream>>>(idx4, D4, H4, W4, D5, H5, W5,
                                                3, 1, 1, 2, 1, 1, 0, 0, 0,
                                                idx5, coords5, cnt + 4);
    build_nbr<<<blks(V5), TB, 0, stream>>>(coords5, cnt + 4, idx4, D4, H4, W4, H5, W5,
                                           3, 1, 1, 2, 1, 1, 0, 0, 0, nbr);
    conv(f4a, 14, nullptr, f5, cnt + 4, V5, 128, 128, 3);

    scatter_output<<<V5, 128, 0, stream>>>(f5, coords5, cnt + 4, (float*)d_out);
}